// UpsampleUpFIRDn_11347303596090
// MI455X (gfx1250) — compile-verified
//
#include <hip/hip_runtime.h>

// Separable 2x upsample with (1,3,3,1)/8 blur kernel, scaled by factor^2:
//   per-axis even phase: 0.25*x[i-1] + 0.75*x[i]
//   per-axis odd  phase: 0.75*x[i]   + 0.25*x[i+1]
// Input  (16,512,64,64) f32  -> Output (16,512,128,128) f32.
// Memory-bound (AI ~1.7 FLOP/B): optimize the data path, not the math.

typedef float v4f __attribute__((ext_vector_type(4)));

#define TS 72          // LDS tile row stride in floats (keeps b128 alignment, 288B rows)
#define TROWS 66       // 64 interior rows + 1 halo row top/bottom

__global__ __launch_bounds__(256)
void upfirdn_up2_k4_kernel(const float* __restrict__ x, float* __restrict__ out) {
  __shared__ float tile[TROWS * TS];

  const int tid = threadIdx.x;                 // 0..255 (8 waves of 32)
  const size_t img = blockIdx.x;               // one (b,c) image per block
  const float* src = x + img * (size_t)4096;   // 64*64
  float* dst = out + img * (size_t)16384;      // 128*128

  // ---- zero the halo ring (input taps outside the image are zero) ----
  // interior lives at rows 1..64, cols 4..67; halo = row 0, row 65, col 3, col 68
  if (tid < 66) {
    tile[0 * TS + 3 + tid]  = 0.0f;
    tile[65 * TS + 3 + tid] = 0.0f;
  }
  if (tid < 64) {
    tile[(1 + tid) * TS + 3]  = 0.0f;
    tile[(1 + tid) * TS + 68] = 0.0f;
  }

  // ---- async global -> LDS copy of the 64x64 interior (ASYNCcnt path) ----
  // GVS mode: SGPR pair = image base, VGPR = 32-bit byte offset, 16B per lane.
  const unsigned long long gbase = (unsigned long long)(size_t)src;
  #pragma unroll
  for (int itc = 0; itc < 4; ++itc) {
    const int f = itc * 1024 + tid * 4;        // flat float index in image
    const int r = f >> 6;                      // input row
    const int c = f & 63;                      // input col (multiple of 4)
    const unsigned lds = (unsigned)(size_t)&tile[(r + 1) * TS + 4 + c]; // 16B aligned
    const unsigned gof = (unsigned)(f * 4);                              // 16B aligned
    asm volatile("global_load_async_to_lds_b128 %0, %1, %2"
                 :: "v"(lds), "v"(gof), "s"(gbase)
                 : "memory");
  }
  asm volatile("s_wait_asynccnt 0x0" ::: "memory");
  __syncthreads();

  // ---- compute: each thread -> 2 output rows x 8 output cols per iteration ----
  const int jg = tid & 15;        // column group: 4 input cols -> 8 output cols
  const int i0 = tid >> 4;        // input row 0..15, stride 16 over 4 iterations
  const int cb = 4 * jg;          // input col base

  #pragma unroll
  for (int ib = 0; ib < 4; ++ib) {
    const int i = i0 + ib * 16;   // input cell row 0..63

    // x[i-1] -> tile row i ; x[i] -> row i+1 ; x[i+1] -> row i+2
    const float* pA = &tile[(i    ) * TS + 4 + cb];
    const float* pB = &tile[(i + 1) * TS + 4 + cb];
    const float* pC = &tile[(i + 2) * TS + 4 + cb];

    const v4f  a = *(const v4f*)pA;  const float aL = pA[-1], aR = pA[4];
    const v4f  b = *(const v4f*)pB;  const float bL = pB[-1], bR = pB[4];
    const v4f  c = *(const v4f*)pC;  const float cL = pC[-1], cR = pC[4];

    // vertical 2-tap filters, 6 columns (cb-1 .. cb+4)
    const float vE0 = 0.25f*aL  + 0.75f*bL;
    const float vE1 = 0.25f*a.x + 0.75f*b.x;
    const float vE2 = 0.25f*a.y + 0.75f*b.y;
    const float vE3 = 0.25f*a.z + 0.75f*b.z;
    const float vE4 = 0.25f*a.w + 0.75f*b.w;
    const float vE5 = 0.25f*aR  + 0.75f*bR;

    const float vO0 = 0.75f*bL  + 0.25f*cL;
    const float vO1 = 0.75f*b.x + 0.25f*c.x;
    const float vO2 = 0.75f*b.y + 0.25f*c.y;
    const float vO3 = 0.75f*b.z + 0.25f*c.z;
    const float vO4 = 0.75f*b.w + 0.25f*c.w;
    const float vO5 = 0.75f*bR  + 0.25f*cR;

    // horizontal 2-tap filters -> interleaved even/odd output columns
    v4f q0, q1, r0, r1;
    q0.x = 0.25f*vE0 + 0.75f*vE1;  q0.y = 0.75f*vE1 + 0.25f*vE2;
    q0.z = 0.25f*vE1 + 0.75f*vE2;  q0.w = 0.75f*vE2 + 0.25f*vE3;
    q1.x = 0.25f*vE2 + 0.75f*vE3;  q1.y = 0.75f*vE3 + 0.25f*vE4;
    q1.z = 0.25f*vE3 + 0.75f*vE4;  q1.w = 0.75f*vE4 + 0.25f*vE5;

    r0.x = 0.25f*vO0 + 0.75f*vO1;  r0.y = 0.75f*vO1 + 0.25f*vO2;
    r0.z = 0.25f*vO1 + 0.75f*vO2;  r0.w = 0.75f*vO2 + 0.25f*vO3;
    r1.x = 0.25f*vO2 + 0.75f*vO3;  r1.y = 0.75f*vO3 + 0.25f*vO4;
    r1.z = 0.25f*vO3 + 0.75f*vO4;  r1.w = 0.75f*vO4 + 0.25f*vO5;

    // streaming (non-temporal) b128 stores; lanes jg=0..15 are contiguous
    float* d0 = dst + (size_t)(2 * i) * 128 + 8 * jg;
    float* d1 = d0 + 128;
    __builtin_nontemporal_store(q0, (v4f*)d0);
    __builtin_nontemporal_store(q1, (v4f*)(d0 + 4));
    __builtin_nontemporal_store(r0, (v4f*)d1);
    __builtin_nontemporal_store(r1, (v4f*)(d1 + 4));
  }
}

extern "C" void kernel_launch(void* const* d_in, const int* in_sizes, int n_in,
                              void* d_out, int out_size, void* d_ws, size_t ws_size,
                              hipStream_t stream) {
  (void)n_in; (void)out_size; (void)d_ws; (void)ws_size;
  const float* x = (const float*)d_in[0];
  float* out = (float*)d_out;
  const int images = in_sizes[0] / 4096;       // (B*C) images of 64x64
  if (images <= 0) return;
  upfirdn_up2_k4_kernel<<<dim3(images), dim3(256), 0, stream>>>(x, out);
}